// SLogSlaterDeterminant_39883066311000
// MI455X (gfx1250) — compile-verified
//
#include <hip/hip_runtime.h>
#include <cmath>

// ---------------------------------------------------------------------------
// SLogSlaterDeterminant for MI455X (gfx1250, wave32).
//
// Roofline: 2048 FLOP vs 1024 B read per batch element -> ~2 FLOP/B, pure
// HBM-bound (~268 MB total -> ~11.5 us floor at 23.3 TB/s). Strategy:
//  * stream input once with full 128B-line utilization,
//  * batched 4x64 @ 64x4 via V_WMMA_F32_16X16X4_F32 (4 batch mats per wave
//    tile: M=16 rows = 4 batches x 4 Slater rows, N=4 live cols, K=64 as
//    16 chained K=4 WMMAs),
//  * weight/bias fragments hoisted to registers (weight is only 256 floats),
//  * 4x4 slogdet via unrolled LU w/ partial pivoting on lanes 0..3,
//    gathered through a per-wave LDS slab (intra-wave ordering only).
// ---------------------------------------------------------------------------

typedef __attribute__((ext_vector_type(2))) float v2f;
typedef __attribute__((ext_vector_type(8))) float v8f;

#define NUM_INPUTS 4
#define IN_FEAT 64
#define HALF_LOG_FACT 1.5890269151739729f   // 0.5 * log(4!)

__global__ __launch_bounds__(256) void slog_slater_wmma_kernel(
    const float* __restrict__ inp,     // [B, 4, 64]
    const float* __restrict__ weight,  // [64, 4]
    const float* __restrict__ bias,    // [4]
    float* __restrict__ out_sign,      // [B]
    float* __restrict__ out_log,       // [B]
    int batch)
{
    __shared__ float lds[8][64];           // one 16x4 D slab per wave

    const int lane = threadIdx.x & 31;
    const int wid  = threadIdx.x >> 5;
    const int m    = lane & 15;            // N column of D (= weight col m)
    const int half = lane >> 4;            // A-lane K group: 0 -> K{0,1}, 1 -> K{2,3}
    const int koff = half * 2;
    const bool mvalid = (m < NUM_INPUTS);

    // ---- B fragments (weight, 64x4) + bias: load once, live in registers ----
    // Guessed f32 B layout (4x16 KxN), mirroring the A layout transposed:
    // VGPR0: K = 4kc+koff, VGPR1: K = 4kc+koff+1, lane%16 = N. Cols >=4 zero.
    v2f bw[16];
#pragma unroll
    for (int kc = 0; kc < 16; ++kc) {
        float b0 = 0.f, b1 = 0.f;
        if (mvalid) {
            b0 = weight[(4 * kc + koff + 0) * NUM_INPUTS + m];
            b1 = weight[(4 * kc + koff + 1) * NUM_INPUTS + m];
        }
        bw[kc].x = b0;
        bw[kc].y = b1;
    }
    const float bm = mvalid ? bias[m] : 0.f;

    const int groups = batch >> 2;              // 4 batch elements per group
    const int gw     = blockIdx.x * 8 + wid;    // global wave id
    const int stride = gridDim.x * 8;

    for (int g = gw; g < groups; g += stride) {
        // Group = 4 consecutive batch elements = 4 KB contiguous input.
        // Row M = 4*b_local + n -> flat offset M*64 inside the group.
        const float* src = inp + (size_t)g * (4 * NUM_INPUTS * IN_FEAT);
        const float* row = src + m * IN_FEAT + koff;

        // A fragments: one global_load_b64 per K-chunk; the 16 unrolled loads
        // cover the wave's 4 KB group with full cache-line utilization.
        v2f a[16];
#pragma unroll
        for (int kc = 0; kc < 16; ++kc)
            a[kc] = *(const v2f*)(row + kc * 4);

        // K=64 accumulation: 16 chained v_wmma_f32_16x16x4_f32
        v8f c = {};
#pragma unroll
        for (int kc = 0; kc < 16; ++kc)
            c = __builtin_amdgcn_wmma_f32_16x16x4_f32(
                    /*neg_a=*/false, a[kc], /*neg_b=*/false, bw[kc],
                    /*c_mod=*/(short)0, c, /*reuse_a=*/false, /*reuse_b=*/false);

        // Scatter D (+bias) to LDS. D layout: VGPR r holds M = r + 8*half,
        // lane%16 = N. Only N < 4 is live.
        if (mvalid) {
#pragma unroll
            for (int r = 0; r < 8; ++r)
                lds[wid][(8 * half + r) * NUM_INPUTS + m] = c[r] + bm;
        }
        // Intra-wave only: LDS ops are in-order per wave; fence keeps the
        // compiler from hoisting the reads and inserts the ds-cnt wait.
        __builtin_amdgcn_fence(__ATOMIC_ACQ_REL, "workgroup");
        __builtin_amdgcn_wave_barrier();

        if (lane < 4) {
            // lane t owns batch element (4g + t): rows 4t..4t+3, cols 0..3
            float A[4][4];
#pragma unroll
            for (int i = 0; i < 4; ++i)
#pragma unroll
                for (int j = 0; j < 4; ++j)
                    A[i][j] = lds[wid][(4 * lane + i) * 4 + j];

            // Unrolled LU with partial pivoting; static-index conditional
            // swaps (pure VALU selects, no dynamic register indexing).
            float sign = 1.f, logabs = 0.f;
#pragma unroll
            for (int k = 0; k < 4; ++k) {
#pragma unroll
                for (int i = k + 1; i < 4; ++i) {
                    bool sw = fabsf(A[i][k]) > fabsf(A[k][k]);
                    if (sw) {
#pragma unroll
                        for (int j = 0; j < 4; ++j) {
                            float t = A[k][j]; A[k][j] = A[i][j]; A[i][j] = t;
                        }
                        sign = -sign;
                    }
                }
                float piv = A[k][k];
                sign   = (piv > 0.f) ? sign : ((piv < 0.f) ? -sign : 0.f);
                logabs += __logf(fabsf(piv));           // log(0) -> -inf (singular)
                float inv = (piv != 0.f) ? (1.f / piv) : 0.f;
#pragma unroll
                for (int i = k + 1; i < 4; ++i) {
                    float f = A[i][k] * inv;
#pragma unroll
                    for (int j = k + 1; j < 4; ++j)
                        A[i][j] -= f * A[k][j];
                }
            }

            int b = g * 4 + lane;
            out_sign[b] = sign;
            out_log[b]  = logabs - HALF_LOG_FACT;
        }

        // WAR ordering before next iteration overwrites the LDS slab.
        __builtin_amdgcn_wave_barrier();
        __builtin_amdgcn_fence(__ATOMIC_ACQ_REL, "workgroup");
    }
}

extern "C" void kernel_launch(void* const* d_in, const int* in_sizes, int n_in,
                              void* d_out, int out_size, void* d_ws, size_t ws_size,
                              hipStream_t stream) {
    const float* inp    = (const float*)d_in[0];   // [B, 4, 64] f32
    const float* weight = (const float*)d_in[1];   // [64, 4]    f32
    const float* bias   = (const float*)d_in[2];   // [4]        f32
    float* out = (float*)d_out;                    // [sign: B][logabsdet: B]

    const int batch  = in_sizes[0] / (NUM_INPUTS * IN_FEAT);
    const int groups = batch / 4;                  // 4 batch elements / wave-tile
    // 8 groups per wave, 8 waves (256 threads) per block.
    int total_waves = (groups + 7) / 8;
    int blocks      = (total_waves + 7) / 8;
    if (blocks < 1) blocks = 1;

    slog_slater_wmma_kernel<<<blocks, 256, 0, stream>>>(
        inp, weight, bias, out, out + batch, batch);
}